// KGINGraphConv_51316269253371
// MI455X (gfx1250) — compile-verified
//
#include <hip/hip_runtime.h>
#include <hip/hip_bf16.h>
#include <math.h>

typedef __attribute__((ext_vector_type(2))) float v2f;
typedef __attribute__((ext_vector_type(8))) float v8f;

#define EMB_D 64
#define NFAC 4
#define NREL 20
#define TEMP 0.2f

__device__ inline float waveReduceSum(float v) {
#pragma unroll
  for (int off = 16; off > 0; off >>= 1)
    v += __shfl_xor(v, off, 32);
  return v;
}

// One block of 256 threads: computes cor scalar, disen = softmax(dwa)@rel,
// and M = latent^T @ disen  ([64x64]) used for the fused mix GEMM.
__global__ void prep_kernel(const float* __restrict__ dwa,   // [F,R]
                            const float* __restrict__ rel,   // [R,D]
                            const float* __restrict__ lat,   // [F,D]
                            float* __restrict__ Mmat,        // [D,D]
                            float* __restrict__ cor_out) {
  __shared__ float dwa_s[NFAC][NREL];
  __shared__ float w_s[NFAC][NREL];
  __shared__ float sm_s[NFAC][NREL];
  __shared__ float disen_s[NFAC][EMB_D];
  int t = threadIdx.x;
  if (t < NFAC * NREL) dwa_s[t / NREL][t % NREL] = dwa[t];
  __syncthreads();
  if (t < NFAC) {
    float ss = 0.f;
    for (int r = 0; r < NREL; ++r) { float x = dwa_s[t][r]; ss += x * x; }
    float inv = 1.f / fmaxf(sqrtf(ss), 1e-12f);
    float mx = -1e30f;
    for (int r = 0; r < NREL; ++r) {
      w_s[t][r] = dwa_s[t][r] * inv;
      mx = fmaxf(mx, dwa_s[t][r]);
    }
    float se = 0.f;
    for (int r = 0; r < NREL; ++r) { float e = expf(dwa_s[t][r] - mx); sm_s[t][r] = e; se += e; }
    float isd = 1.f / se;
    for (int r = 0; r < NREL; ++r) sm_s[t][r] *= isd;
  }
  __syncthreads();
  if (t == 0) {
    float cor = 0.f;
    for (int i = 0; i < NFAC; ++i) {
      float lg[NFAC];
      float mx = -1e30f;
      for (int j = 0; j < NFAC; ++j) {
        float dv = 0.f;
        for (int r = 0; r < NREL; ++r) dv += w_s[i][r] * w_s[j][r];
        lg[j] = dv / TEMP;
        mx = fmaxf(mx, lg[j]);
      }
      float se = 0.f;
      for (int j = 0; j < NFAC; ++j) se += expf(lg[j] - mx);
      float lse = mx + logf(se);
      cor -= (lg[i] - lse);
    }
    *cor_out = cor;
  }
  {
    int f = t / EMB_D, d = t % EMB_D;  // blockDim == 256 == F*D
    float acc = 0.f;
    for (int r = 0; r < NREL; ++r) acc += sm_s[f][r] * rel[r * EMB_D + d];
    disen_s[f][d] = acc;
  }
  __syncthreads();
  for (int idx = t; idx < EMB_D * EMB_D; idx += 256) {
    int d1 = idx / EMB_D, d2 = idx % EMB_D;
    float acc = 0.f;
#pragma unroll
    for (int f = 0; f < NFAC; ++f) acc += lat[f * EMB_D + d1] * disen_s[f][d2];
    Mmat[idx] = acc;
  }
}

__global__ void deg_count_kernel(const int* __restrict__ dst, float* __restrict__ deg, int E) {
  int i = blockIdx.x * blockDim.x + threadIdx.x;
  if (i < E) atomicAdd(&deg[dst[i]], 1.0f);
}

__global__ void inv_deg_kernel(float* __restrict__ deg, int N) {
  int i = blockIdx.x * blockDim.x + threadIdx.x;
  if (i < N) { float d = deg[i]; deg[i] = d > 0.f ? 1.f / d : 0.f; }
}

// 16 threads per edge, float4 per thread. msg = rel_emb[rel] * ent[src] -> atomicAdd neigh[dst]
__global__ void kg_scatter_kernel(const int* __restrict__ src, const int* __restrict__ dst,
                                  const int* __restrict__ erel,
                                  const float* __restrict__ rel,  // [R,D]
                                  const float* __restrict__ ent,  // [N,D]
                                  float* __restrict__ neigh, int E) {
  long g = (long)blockIdx.x * blockDim.x + threadIdx.x;
  long e = g >> 4;
  int q = (int)(g & 15);
  if (e >= E) return;
  int s = src[e], d = dst[e], r = erel[e];
  const float4 rv = ((const float4*)rel)[r * 16 + q];
  const float4 sv = ((const float4*)ent)[(long)s * 16 + q];
  float* out = neigh + (long)d * EMB_D + q * 4;
  atomicAdd(out + 0, rv.x * sv.x);
  atomicAdd(out + 1, rv.y * sv.y);
  atomicAdd(out + 2, rv.z * sv.z);
  atomicAdd(out + 3, rv.w * sv.w);
}

// vals = inter_values * ent[item] -> atomicAdd user_agg[user]
__global__ void user_scatter_kernel(const int* __restrict__ iu, const int* __restrict__ ii,
                                    const float* __restrict__ vals,
                                    const float* __restrict__ ent,
                                    float* __restrict__ agg, int E) {
  long g = (long)blockIdx.x * blockDim.x + threadIdx.x;
  long e = g >> 4;
  int q = (int)(g & 15);
  if (e >= E) return;
  int u = iu[e], it = ii[e];
  float v = vals[e];
  const float4 sv = ((const float4*)ent)[(long)it * 16 + q];
  float* out = agg + (long)u * EMB_D + q * 4;
  atomicAdd(out + 0, v * sv.x);
  atomicAdd(out + 1, v * sv.y);
  atomicAdd(out + 2, v * sv.z);
  atomicAdd(out + 3, v * sv.w);
}

// One wave per 16-user tile: mix = usr @ M via V_WMMA_F32_16X16X4_F32,
// then agg *= (1 + mix) applied in place from the known C-matrix layout.
__global__ void __launch_bounds__(32)
wmma_mix_kernel(const float* __restrict__ usr, const float* __restrict__ Mmat,
                float* __restrict__ agg, int NU) {
  int u0 = blockIdx.x * 16;
  int lane = threadIdx.x;
  if (u0 + 16 <= NU) {
    v8f c0 = {}, c1 = {}, c2 = {}, c3 = {};
    int m = lane & 15;
    int khalf = lane >> 4;  // A/B layout: lanes 0-15 hold K0,K1; lanes 16-31 hold K2,K3
    const float* arow = usr + (long)(u0 + m) * EMB_D;
#pragma unroll
    for (int ks = 0; ks < 16; ++ks) {
      int k0 = ks * 4 + khalf * 2;
      v2f a;
      a.x = arow[k0];
      a.y = arow[k0 + 1];
      v2f b0, b1, b2, b3;
      b0.x = Mmat[(k0 + 0) * EMB_D + 0 * 16 + m];  b0.y = Mmat[(k0 + 1) * EMB_D + 0 * 16 + m];
      b1.x = Mmat[(k0 + 0) * EMB_D + 1 * 16 + m];  b1.y = Mmat[(k0 + 1) * EMB_D + 1 * 16 + m];
      b2.x = Mmat[(k0 + 0) * EMB_D + 2 * 16 + m];  b2.y = Mmat[(k0 + 1) * EMB_D + 2 * 16 + m];
      b3.x = Mmat[(k0 + 0) * EMB_D + 3 * 16 + m];  b3.y = Mmat[(k0 + 1) * EMB_D + 3 * 16 + m];
      c0 = __builtin_amdgcn_wmma_f32_16x16x4_f32(false, a, false, b0, (short)0, c0, false, false);
      c1 = __builtin_amdgcn_wmma_f32_16x16x4_f32(false, a, false, b1, (short)0, c1, false, false);
      c2 = __builtin_amdgcn_wmma_f32_16x16x4_f32(false, a, false, b2, (short)0, c2, false, false);
      c3 = __builtin_amdgcn_wmma_f32_16x16x4_f32(false, a, false, b3, (short)0, c3, false, false);
    }
    // C layout: VGPR j -> row m=j (lanes 0-15) / m=j+8 (lanes 16-31), N = lane%16
    int rbase = u0 + ((lane >> 4) << 3);
    int col = lane & 15;
#pragma unroll
    for (int j = 0; j < 8; ++j) {
      long row = rbase + j;
      float* p0 = agg + row * EMB_D + 0 * 16 + col;
      float* p1 = agg + row * EMB_D + 1 * 16 + col;
      float* p2 = agg + row * EMB_D + 2 * 16 + col;
      float* p3 = agg + row * EMB_D + 3 * 16 + col;
      *p0 = *p0 * (1.f + c0[j]);
      *p1 = *p1 * (1.f + c1[j]);
      *p2 = *p2 * (1.f + c2[j]);
      *p3 = *p3 * (1.f + c3[j]);
    }
  } else {
    // scalar tail (block-uniform branch; never taken when NU % 16 == 0)
    for (int idx = lane; idx < 16 * EMB_D; idx += 32) {
      int r = idx >> 6, col = idx & 63;
      int row = u0 + r;
      if (row < NU) {
        const float* arow = usr + (long)row * EMB_D;
        float acc = 0.f;
        for (int k = 0; k < EMB_D; ++k) acc += arow[k] * Mmat[k * EMB_D + col];
        float* p = agg + (long)row * EMB_D + col;
        *p = *p * (1.f + acc);
      }
    }
  }
}

// One wave per row: optional inv_deg scale, l2norm in place, residual into d_out.
__global__ void norm_residual_kernel(float* __restrict__ buf, const float* __restrict__ invdeg,
                                     const float* __restrict__ base, float* __restrict__ res,
                                     int N, int first) {
  int wid = threadIdx.x >> 5;
  int lane = threadIdx.x & 31;
  long row = (long)blockIdx.x * 8 + wid;
  if (row >= N) return;
  float2* p = (float2*)(buf + row * EMB_D) + lane;
  float2 v = *p;
  if (invdeg) { float s = invdeg[row]; v.x *= s; v.y *= s; }
  float ss = waveReduceSum(v.x * v.x + v.y * v.y);
  float sc = 1.f / fmaxf(sqrtf(ss), 1e-12f);
  v.x *= sc;
  v.y *= sc;
  *p = v;
  float2* rp = (float2*)(res + row * EMB_D) + lane;
  if (first) {
    const float2* bp = (const float2*)(base + row * EMB_D) + lane;
    float2 b = *bp;
    float2 o;
    o.x = b.x + v.x;
    o.y = b.y + v.y;
    *rp = o;
  } else {
    float2 old = *rp;
    old.x += v.x;
    old.y += v.y;
    *rp = old;
  }
}

extern "C" void kernel_launch(void* const* d_in, const int* in_sizes, int n_in,
                              void* d_out, int out_size, void* d_ws, size_t ws_size,
                              hipStream_t stream) {
  const float* user_emb = (const float*)d_in[0];
  const float* ent_emb = (const float*)d_in[1];
  const float* lat_emb = (const float*)d_in[2];
  const float* rel_emb = (const float*)d_in[3];
  const float* dwa = (const float*)d_in[4];
  const float* inter_val = (const float*)d_in[5];
  const int* edge_src = (const int*)d_in[6];
  const int* edge_dst = (const int*)d_in[7];
  const int* edge_rel = (const int*)d_in[8];
  const int* inter_user = (const int*)d_in[9];
  const int* inter_item = (const int*)d_in[10];

  const int NU = in_sizes[0] / EMB_D;
  const int NE = in_sizes[1] / EMB_D;
  const int EK = in_sizes[6];
  const int EI = in_sizes[5];

  float* out = (float*)d_out;
  float* ent_res = out;
  float* user_res = out + (long)NE * EMB_D;
  float* cor_out = out + (out_size - 1);

  char* ws = (char*)d_ws;
  auto al = [](size_t x) { return (x + 255) & ~(size_t)255; };
  size_t off = 0;
  float* Mmat = (float*)(ws + off); off = al(off + (size_t)EMB_D * EMB_D * sizeof(float));
  float* deg  = (float*)(ws + off); off = al(off + (size_t)NE * sizeof(float));
  float* eA   = (float*)(ws + off); off = al(off + (size_t)NE * EMB_D * sizeof(float));
  float* eB   = (float*)(ws + off); off = al(off + (size_t)NE * EMB_D * sizeof(float));
  float* uA   = (float*)(ws + off); off = al(off + (size_t)NU * EMB_D * sizeof(float));
  float* uB   = (float*)(ws + off); off = al(off + (size_t)NU * EMB_D * sizeof(float));
  (void)ws_size; (void)n_in;

  prep_kernel<<<1, 256, 0, stream>>>(dwa, rel_emb, lat_emb, Mmat, cor_out);

  hipMemsetAsync(deg, 0, (size_t)NE * sizeof(float), stream);
  deg_count_kernel<<<(EK + 255) / 256, 256, 0, stream>>>(edge_dst, deg, EK);
  inv_deg_kernel<<<(NE + 255) / 256, 256, 0, stream>>>(deg, NE);

  const float* ent = ent_emb;
  const float* usr = user_emb;
  float* ebuf[2] = {eA, eB};
  float* ubuf[2] = {uA, uB};
  for (int layer = 0; layer < 3; ++layer) {
    float* neigh = ebuf[layer & 1];
    float* agg = ubuf[layer & 1];
    hipMemsetAsync(neigh, 0, (size_t)NE * EMB_D * sizeof(float), stream);
    hipMemsetAsync(agg, 0, (size_t)NU * EMB_D * sizeof(float), stream);
    {
      long tot = (long)EK * 16;
      kg_scatter_kernel<<<(int)((tot + 255) / 256), 256, 0, stream>>>(
          edge_src, edge_dst, edge_rel, rel_emb, ent, neigh, EK);
    }
    {
      long tot = (long)EI * 16;
      user_scatter_kernel<<<(int)((tot + 255) / 256), 256, 0, stream>>>(
          inter_user, inter_item, inter_val, ent, agg, EI);
    }
    wmma_mix_kernel<<<(NU + 15) / 16, 32, 0, stream>>>(usr, Mmat, agg, NU);
    norm_residual_kernel<<<(NE + 7) / 8, 256, 0, stream>>>(neigh, deg, ent_emb, ent_res, NE,
                                                           layer == 0 ? 1 : 0);
    norm_residual_kernel<<<(NU + 7) / 8, 256, 0, stream>>>(agg, nullptr, user_emb, user_res, NU,
                                                           layer == 0 ? 1 : 0);
    ent = neigh;
    usr = agg;
  }
}